// DirectionalTransform_59459527245955
// MI455X (gfx1250) — compile-verified
//
#include <hip/hip_runtime.h>
#include <math.h>

// Bicubic grid_sample (zeros padding, align_corners=False), N=1.
// Memory-bound: ~300MB traffic -> ~13us floor at 23.3 TB/s HBM.
// Strategy:
//  - x (134MB) kept resident in the 192MB L2; grid/out streamed with
//    non-temporal hints so they do not evict x.
//  - Interior fast path: the 4 contiguous x-taps are covered by two ALIGNED
//    global_load_b128's ([a, a+7], a = (ix0-1)&~3 — same cache lines as the
//    unaligned span), dotted with a shift-padded weight vector e[0..6] built
//    once per pixel. Halves VMEM instruction count vs 4x b32 gathers.
//  - Rim path: per-tap masks + clamped indices (zeros padding).

#define A_ (-0.75f)

typedef float v2f __attribute__((ext_vector_type(2)));
typedef float v4f __attribute__((ext_vector_type(4)));

__device__ __forceinline__ void cubic_w(float t, float w[4]) {
    float s0 = t + 1.0f;
    w[0] = ((A_ * s0 - 5.0f * A_) * s0 + 8.0f * A_) * s0 - 4.0f * A_;
    w[1] = ((A_ + 2.0f) * t - (A_ + 3.0f)) * t * t + 1.0f;
    float u = 1.0f - t;
    w[2] = ((A_ + 2.0f) * u - (A_ + 3.0f)) * u * u + 1.0f;
    float s3 = 2.0f - t;
    w[3] = ((A_ * s3 - 5.0f * A_) * s3 + 8.0f * A_) * s3 - 4.0f * A_;
}

template <int C>
__global__ __launch_bounds__(256) void bicubic_gs_kernel(
    const float* __restrict__ x,     // [C,H,W]
    const float* __restrict__ grid,  // [H,W,2]
    float* __restrict__ out,         // [C,H,W]
    int H, int W) {
    const long long HW  = (long long)H * (long long)W;
    const long long pix = (long long)blockIdx.x * blockDim.x + threadIdx.x;
    if (pix >= HW) return;

    // grid streams linearly: non-temporal load + gfx1250 prefetch ahead
    const v2f* gp = (const v2f*)grid;
    __builtin_prefetch((const void*)(gp + pix + 2048), 0, 1);
    v2f g = __builtin_nontemporal_load(gp + pix);

    // unnormalize (align_corners=False)
    const float ix = ((g.x + 1.0f) * (float)W - 1.0f) * 0.5f;
    const float iy = ((g.y + 1.0f) * (float)H - 1.0f) * 0.5f;
    const float ix0f = floorf(ix);
    const float iy0f = floorf(iy);
    const int ix0 = (int)ix0f;
    const int iy0 = (int)iy0f;

    float wx[4], wy[4];
    cubic_w(ix - ix0f, wx);
    cubic_w(iy - iy0f, wy);

    float acc[C];
#pragma unroll
    for (int c = 0; c < C; ++c) acc[c] = 0.0f;

    // iy0 <= H-4 keeps the widened (a..a+7) read of the bottom row inside the
    // tensor even for the last channel.
    const bool interior = (ix0 >= 1) & (ix0 <= W - 3) & (iy0 >= 1) & (iy0 <= H - 4);

    if (interior) {
        const int p   = ix0 - 1;      // first tap column
        const int a   = p & ~3;       // 16B-aligned window start
        const int off = p & 3;        // tap shift within window

        // Shift-padded weight vector: e[j] = wx[j-off] for j in [off, off+3], else 0.
        const float w0 = wx[0], w1 = wx[1], w2 = wx[2], w3 = wx[3];
        const float e0 = (off == 0) ? w0 : 0.0f;
        const float e1 = (off == 0) ? w1 : (off == 1) ? w0 : 0.0f;
        const float e2 = (off == 0) ? w2 : (off == 1) ? w1 : (off == 2) ? w0 : 0.0f;
        const float e3 = (off == 0) ? w3 : (off == 1) ? w2 : (off == 2) ? w1 : w0;
        const float e4 = (off == 1) ? w3 : (off == 2) ? w2 : (off == 3) ? w1 : 0.0f;
        const float e5 = (off == 2) ? w3 : (off == 3) ? w2 : 0.0f;
        const float e6 = (off == 3) ? w3 : 0.0f;

        // 32-bit element offset (max tensor 33.5M elements < 2^32)
        const unsigned rowbase = (unsigned)(iy0 - 1) * (unsigned)W + (unsigned)a;

#pragma unroll
        for (int dy = 0; dy < 4; ++dy) {
            const unsigned ro = rowbase + (unsigned)dy * (unsigned)W;
            const float wyv = wy[dy];
#pragma unroll
            for (int c = 0; c < C; ++c) {
                const float* q = x + (long long)c * HW + ro;
                v4f lo = *(const v4f*)q;        // aligned global_load_b128
                v4f hi = *(const v4f*)(q + 4);  // aligned global_load_b128
                float s = fmaf(lo.x, e0,
                          fmaf(lo.y, e1,
                          fmaf(lo.z, e2,
                          fmaf(lo.w, e3,
                          fmaf(hi.x, e4,
                          fmaf(hi.y, e5, hi.z * e6))))));
                acc[c] = fmaf(wyv, s, acc[c]);
            }
        }
    } else {
        // Rim path: zeros padding via masks, clamped gather indices.
#pragma unroll
        for (int dy = 0; dy < 4; ++dy) {
            const int  yi  = iy0 + dy - 1;
            const bool vy  = (yi >= 0) & (yi < H);
            const int  yic = min(max(yi, 0), H - 1);
            const long long rowOff = (long long)yic * W;
#pragma unroll
            for (int dx = 0; dx < 4; ++dx) {
                const int  xi  = ix0 + dx - 1;
                const bool m   = vy & (xi >= 0) & (xi < W);
                const int  xic = min(max(xi, 0), W - 1);
                const float w  = wy[dy] * wx[dx] * (m ? 1.0f : 0.0f);
                const float* p = x + rowOff + xic;
#pragma unroll
                for (int c = 0; c < C; ++c)
                    acc[c] = fmaf(p[(long long)c * HW], w, acc[c]);
            }
        }
    }

    // out streams once: non-temporal stores keep x resident in L2
#pragma unroll
    for (int c = 0; c < C; ++c)
        __builtin_nontemporal_store(acc[c], out + (long long)c * HW + pix);
}

// Generic fallback (runtime channel count / odd widths), scalar gathers.
__global__ __launch_bounds__(256) void bicubic_gs_kernel_dyn(
    const float* __restrict__ x, const float* __restrict__ grid,
    float* __restrict__ out, int C, int H, int W) {
    const long long HW  = (long long)H * (long long)W;
    const long long pix = (long long)blockIdx.x * blockDim.x + threadIdx.x;
    if (pix >= HW) return;

    const v2f* gp = (const v2f*)grid;
    v2f g = __builtin_nontemporal_load(gp + pix);
    const float ix = ((g.x + 1.0f) * (float)W - 1.0f) * 0.5f;
    const float iy = ((g.y + 1.0f) * (float)H - 1.0f) * 0.5f;
    const float ix0f = floorf(ix), iy0f = floorf(iy);
    const int ix0 = (int)ix0f, iy0 = (int)iy0f;
    float wx[4], wy[4];
    cubic_w(ix - ix0f, wx);
    cubic_w(iy - iy0f, wy);

    for (int c = 0; c < C; ++c) {
        float acc = 0.0f;
        for (int dy = 0; dy < 4; ++dy) {
            const int  yi  = iy0 + dy - 1;
            const bool vy  = (yi >= 0) & (yi < H);
            const int  yic = min(max(yi, 0), H - 1);
            for (int dx = 0; dx < 4; ++dx) {
                const int  xi  = ix0 + dx - 1;
                const bool m   = vy & (xi >= 0) & (xi < W);
                const int  xic = min(max(xi, 0), W - 1);
                const float w  = wy[dy] * wx[dx] * (m ? 1.0f : 0.0f);
                acc = fmaf(x[(long long)c * HW + (long long)yic * W + xic], w, acc);
            }
        }
        __builtin_nontemporal_store(acc, out + (long long)c * HW + pix);
    }
}

extern "C" void kernel_launch(void* const* d_in, const int* in_sizes, int n_in,
                              void* d_out, int out_size, void* d_ws, size_t ws_size,
                              hipStream_t stream) {
    const float* x    = (const float*)d_in[0];  // [1,C,H,W]
    const float* grid = (const float*)d_in[1];  // [1,H,W,2]
    float* out        = (float*)d_out;          // [1,C,H,W]

    const long long hw = (long long)in_sizes[1] / 2;        // H*W (square grid)
    const int H = (int)(sqrt((double)hw) + 0.5);
    const int W = H;
    const int C = (int)((long long)in_sizes[0] / hw);

    const int threads = 256;
    const int blocks  = (int)((hw + threads - 1) / threads);

    if (C == 8 && (W % 4) == 0 && H >= 8 && W >= 8) {
        bicubic_gs_kernel<8><<<blocks, threads, 0, stream>>>(x, grid, out, H, W);
    } else {
        bicubic_gs_kernel_dyn<<<blocks, threads, 0, stream>>>(x, grid, out, C, H, W);
    }
}